// Modified3DFisherVectors_58746562675185
// MI455X (gfx1250) — compile-verified
//
#include <hip/hip_runtime.h>
#include <math.h>

// ---------------------------------------------------------------------------
// Modified 3D Fisher Vectors on MI455X (gfx1250, wave32).
//
// Roofline: 1.5MB of inputs vs ~16.4M point-gaussian pairs of exp-heavy work
// -> pure compute, transcendental-pipe bound. The Mahalanobis+log-weight term
// ln(w_k) - sum(ln sigma) - 1.5 ln(2pi) - 0.5*||(x-mu)/sigma||^2 is a GEMM
// [16 pts x 7] * [7 x 16 gaussians] run on the WMMA pipe
// (V_WMMA_F32_16X16X4_F32, two k=4 slices per 16x16 tile). The gaussian
// coefficient matrix is precomputed once into a 4KB lane-indexed table so the
// WMMA loop is load_b128 + 2 WMMA + 8 exp per tile.
// ---------------------------------------------------------------------------

typedef __attribute__((ext_vector_type(2))) float v2f;
typedef __attribute__((ext_vector_type(8))) float v8f;

#define B_      16
#define N_      8192
#define K_      125
#define KT      8             // 8 tiles of 16 gaussians (padded to 128)
#define NCHUNK  16            // point chunks per batch in the accumulation pass
#define CHUNK   (N_ / NCHUNK) // 512
#define C15     0.06349363593424097f   // (2*pi)^{-1.5}

// ---------------------------------------------------------------------------
// Kernel P: precompute the WMMA B-operand table, one float4 per (tile, lane).
// B-matrix 4x16 f32 layout (ISA 7.12.2 mirror of A 16x4):
//   reg0 = lanes0-15: row j, lanes16-31: row j+2 ; per k-slice.
// Rows j: 0..2 = -0.5/s_d^2 ; 3..5 = mu_d/s_d^2 ;
//         6 = ln(w*(2pi)^-1.5/(s0 s1 s2)) - 0.5*sum(mu^2/s^2) ; 7 = 0.
// Padding columns k>=125 get row6 = -inf  =>  exp() = 0 in the k-sum.
// ---------------------------------------------------------------------------
__global__ __launch_bounds__(256)
void fv_coefs(const float* __restrict__ w, const float* __restrict__ mu,
              const float* __restrict__ sg, float4* __restrict__ c4)
{
    const int tid  = threadIdx.x;        // == tile*32 + lane
    const int lane = tid & 31;
    const int t    = tid >> 5;
    const int m    = lane & 15;
    const bool hi  = lane >= 16;
    const int  kc    = t * 16 + m;
    const bool valid = kc < K_;
    const int  ki    = valid ? kc : 0;

    const float m0 = mu[ki*3+0], m1 = mu[ki*3+1], m2 = mu[ki*3+2];
    const float s0 = sg[ki*3+0], s1 = sg[ki*3+1], s2 = sg[ki*3+2];
    const float i0 = 1.0f/(s0*s0), i1 = 1.0f/(s1*s1), i2 = 1.0f/(s2*s2);
    const float g0 = -0.5f*i0, g1 = -0.5f*i1, g2 = -0.5f*i2;
    const float g3 = m0*i0,    g4 = m1*i1,    g5 = m2*i2;
    const float lq = logf(w[ki] * C15 / (s0 * s1 * s2));
    const float g6 = valid ? (lq - 0.5f*(m0*m0*i0 + m1*m1*i1 + m2*m2*i2))
                           : -INFINITY;

    float4 v;                            // (b1.x, b1.y, b2.x, b2.y)
    v.x = hi ? g2   : g0;                // rows 2 / 0
    v.y = hi ? g3   : g1;                // rows 3 / 1
    v.z = hi ? g6   : g4;                // rows 6 / 4
    v.w = hi ? 0.0f : g5;                // rows 7 / 5
    c4[tid] = v;
}

// ---------------------------------------------------------------------------
// Kernel A: per-point softmax denominators via WMMA.
// One wave handles 16 consecutive points. A-matrix (16x4 f32, 2 VGPRs):
//   reg0 = lanes0-15: j0, lanes16-31: j2 ; reg1 = j1 / j3   (ISA 7.12.2)
// Feature order j: x^2 y^2 z^2 x y z 1 0.
// C layout: lane holds column N = lane%16, rows M = reg + 8*(lane>=16).
// ---------------------------------------------------------------------------
__global__ __launch_bounds__(256)
void fv_denom_wmma(const float* __restrict__ pts,
                   const float4* __restrict__ c4,
                   float* __restrict__ rdenom)
{
    const int lane = threadIdx.x & 31;
    const int wave = threadIdx.x >> 5;
    const int tile = blockIdx.x * 8 + wave;      // 16-point tile id
    const int b    = tile >> 9;                  // / (N_/16 = 512)
    const int n0   = (tile & 511) << 4;
    const int m    = lane & 15;
    const bool hi  = lane >= 16;

    const float* p = pts + (size_t)(b * N_ + n0 + m) * 3;
    const float px = p[0], py = p[1], pz = p[2];

    v2f a1, a2;                                  // A operand, j=0..3 and j=4..7
    a1.x = hi ? pz * pz : px * px;
    a1.y = hi ? px      : py * py;
    a2.x = hi ? 1.0f    : py;
    a2.y = hi ? 0.0f    : pz;

    float acc[8];
#pragma unroll
    for (int r = 0; r < 8; ++r) acc[r] = 0.0f;

#pragma unroll
    for (int t = 0; t < KT; ++t) {
        const float4 bc = c4[t * 32 + lane];     // cached B-operand slice
        v2f b1, b2;
        b1.x = bc.x; b1.y = bc.y;
        b2.x = bc.z; b2.y = bc.w;

        v8f c = {};
        c = __builtin_amdgcn_wmma_f32_16x16x4_f32(false, a1, false, b1,
                                                  (short)0, c, false, false);
        c = __builtin_amdgcn_wmma_f32_16x16x4_f32(false, a2, false, b2,
                                                  (short)0, c, false, false);

        // c[r] = ln p[m_r, kc] ; padding columns are -inf -> exp == 0
#pragma unroll
        for (int r = 0; r < 8; ++r)
            acc[r] += __expf(c[r]);
    }

    // Sum over the 16 gaussian columns held by each 16-lane half.
#pragma unroll
    for (int r = 0; r < 8; ++r) {
        float v = acc[r];
        v += __shfl_xor(v, 1, 32);
        v += __shfl_xor(v, 2, 32);
        v += __shfl_xor(v, 4, 32);
        v += __shfl_xor(v, 8, 32);
        acc[r] = v;
    }
    if (lane == 0) {
#pragma unroll
        for (int r = 0; r < 8; ++r)
            rdenom[b * N_ + n0 + r] = 1.0f / acc[r];
    } else if (lane == 16) {
#pragma unroll
        for (int r = 0; r < 8; ++r)
            rdenom[b * N_ + n0 + 8 + r] = 1.0f / acc[r];
    }
}

// ---------------------------------------------------------------------------
// Kernel B: gaussian-per-thread accumulation of the 20 per-(b,k) statistics
// over a 512-point chunk. Points + reciprocal denominators staged in LDS
// (uniform-address broadcast reads). Scaling deferred: max/min commute with
// the positive per-k scale factors.
// ---------------------------------------------------------------------------
__global__ __launch_bounds__(128)
void fv_accum(const float* __restrict__ pts, const float* __restrict__ w,
              const float* __restrict__ mu, const float* __restrict__ sg,
              const float* __restrict__ rdenom, float* __restrict__ partial)
{
    __shared__ float lx[CHUNK], ly[CHUNK], lz[CHUNK], lrd[CHUNK];
    const int b  = blockIdx.x >> 4;              // NCHUNK == 16
    const int ch = blockIdx.x & 15;
    const int n0 = ch * CHUNK;
    const int t  = threadIdx.x;

    for (int i = t; i < CHUNK; i += 128) {
        const int n = b * N_ + n0 + i;
        const float* p = pts + (size_t)n * 3;
        lx[i] = p[0]; ly[i] = p[1]; lz[i] = p[2];
        lrd[i] = rdenom[n];
    }
    __syncthreads();

    const int kk = (t < K_) ? t : 0;
    const float m0 = mu[kk*3+0], m1 = mu[kk*3+1], m2 = mu[kk*3+2];
    const float s0 = sg[kk*3+0], s1 = sg[kk*3+1], s2 = sg[kk*3+2];
    const float rs0 = 1.0f/s0, rs1 = 1.0f/s1, rs2 = 1.0f/s2;
    const float q = w[kk] * C15 * rs0 * rs1 * rs2;

    float sQ = 0.0f, mQ = -INFINITY;
    float sM[3] = {0,0,0}, xM[3] = {-INFINITY,-INFINITY,-INFINITY};
    float nM[3] = { INFINITY, INFINITY, INFINITY};
    float sS[3] = {0,0,0}, xS[3] = {-INFINITY,-INFINITY,-INFINITY};
    float nS[3] = { INFINITY, INFINITY, INFINITY};

    for (int i = 0; i < CHUNK; ++i) {
        float z0 = (lx[i] - m0) * rs0;
        float z1 = (ly[i] - m1) * rs1;
        float z2 = (lz[i] - m2) * rs2;
        float e  = -0.5f * (z0*z0 + z1*z1 + z2*z2);
        float Q  = q * __expf(e) * lrd[i];
        sQ += Q; mQ = fmaxf(mQ, Q);
        float z[3] = {z0, z1, z2};
#pragma unroll
        for (int d = 0; d < 3; ++d) {
            float u = Q * z[d];
            sM[d] += u; xM[d] = fmaxf(xM[d], u); nM[d] = fminf(nM[d], u);
            float v = Q * (z[d]*z[d] - 1.0f);
            sS[d] += v; xS[d] = fmaxf(xS[d], v); nS[d] = fminf(nS[d], v);
        }
    }

    if (t < K_) {
        float* o = partial + ((size_t)(b * NCHUNK + ch) * K_ + t) * 20;
        o[0] = mQ;  o[1] = sQ;
#pragma unroll
        for (int d = 0; d < 3; ++d) {
            o[2  + d] = xM[d]; o[5  + d] = nM[d]; o[8  + d] = sM[d];
            o[11 + d] = xS[d]; o[14 + d] = nS[d]; o[17 + d] = sS[d];
        }
    }
}

// ---------------------------------------------------------------------------
// Kernel C: deterministic chunk combine, scaling, signed-sqrt power norm,
// per-feature L2 norm over K, transposed [B,20,K] output.
// ---------------------------------------------------------------------------
__global__ __launch_bounds__(128)
void fv_finalize(const float* __restrict__ w, const float* __restrict__ partial,
                 float* __restrict__ out)
{
    __shared__ float vals[20][128];
    __shared__ float ssq[20];
    const int b = blockIdx.x;
    const int t = threadIdx.x;

    float v[20];
    if (t < K_) {
        float a[20];
        {
            const float* p = partial + ((size_t)(b * NCHUNK) * K_ + t) * 20;
#pragma unroll
            for (int f = 0; f < 20; ++f) a[f] = p[f];
        }
        for (int c = 1; c < NCHUNK; ++c) {
            const float* p = partial + ((size_t)(b * NCHUNK + c) * K_ + t) * 20;
            a[0] = fmaxf(a[0], p[0]); a[1] += p[1];
#pragma unroll
            for (int d = 0; d < 3; ++d) {
                a[2  + d] = fmaxf(a[2  + d], p[2  + d]);
                a[5  + d] = fminf(a[5  + d], p[5  + d]);
                a[8  + d] +=            p[8  + d];
                a[11 + d] = fmaxf(a[11 + d], p[11 + d]);
                a[14 + d] = fminf(a[14 + d], p[14 + d]);
                a[17 + d] +=            p[17 + d];
            }
        }
        const float wk  = w[t];
        const float Nf  = (float)N_;
        const float spi = 1.0f / (sqrtf(wk) * Nf);           // also d_mu scale
        const float ssg = 1.0f / (Nf * sqrtf(2.0f * wk));
        v[0] = (a[0] - wk)      * spi;                       // max (Q - w)
        v[1] = (a[1] - Nf * wk) * spi;                       // sum (Q - w)
#pragma unroll
        for (int f = 2;  f < 11; ++f) v[f] = a[f] * spi;
#pragma unroll
        for (int f = 11; f < 20; ++f) v[f] = a[f] * ssg;
#pragma unroll
        for (int f = 0; f < 20; ++f)                         // power norm
            v[f] = copysignf(sqrtf(fabsf(v[f])), v[f]);
    } else {
#pragma unroll
        for (int f = 0; f < 20; ++f) v[f] = 0.0f;            // zero padding
    }
#pragma unroll
    for (int f = 0; f < 20; ++f) vals[f][t] = v[f];
    __syncthreads();

    if (t < 20) {
        float s = 0.0f;
        for (int k = 0; k < K_; ++k) { float x = vals[t][k]; s += x * x; }
        ssq[t] = s;
    }
    __syncthreads();

    if (t < K_) {
#pragma unroll
        for (int f = 0; f < 20; ++f)
            out[(size_t)b * 20 * K_ + f * K_ + t] =
                v[f] * rsqrtf(fmaxf(ssq[f], 1e-12f));
    }
}

// ---------------------------------------------------------------------------
extern "C" void kernel_launch(void* const* d_in, const int* in_sizes, int n_in,
                              void* d_out, int out_size, void* d_ws, size_t ws_size,
                              hipStream_t stream)
{
    const float* pts = (const float*)d_in[0];   // [16, 8192, 3]
    const float* w   = (const float*)d_in[1];   // [125]
    const float* mu  = (const float*)d_in[2];   // [125, 3]
    const float* sg  = (const float*)d_in[3];   // [125, 3]
    float* out = (float*)d_out;                 // [16, 20, 125]

    float*  rdenom  = (float*)d_ws;             // B*N floats        (512 KB)
    float*  partial = rdenom + B_ * N_;         // B*16*125*20 fl    (2.56 MB)
    float4* c4      = (float4*)(partial + (size_t)B_ * NCHUNK * K_ * 20); // 4KB

    fv_coefs<<<1, 256, 0, stream>>>(w, mu, sg, c4);
    // 8192 point-tiles, 8 waves (tiles) per 256-thread block.
    fv_denom_wmma<<<(B_ * N_ / 16) / 8, 256, 0, stream>>>(pts, c4, rdenom);
    fv_accum<<<B_ * NCHUNK, 128, 0, stream>>>(pts, w, mu, sg, rdenom, partial);
    fv_finalize<<<B_, 128, 0, stream>>>(w, partial, out);
}